// OrthogonalPolynomial_47296179863826
// MI455X (gfx1250) — compile-verified
//
#include <hip/hip_runtime.h>

// Orthogonal polynomial recurrence, N=500000 points x P=128 basis functions.
// Output = 256 MB f32 -> pure HBM-write-bound (~11 us floor @ 23.3 TB/s).
// Strategy: one thread per xi, recurrence in registers, outputs staged through
// padded LDS and written back as fully-coalesced 512B nontemporal wave stores.
// Coefficients staged global->LDS via the gfx1250 async-LDS path when available.

typedef float v4f __attribute__((ext_vector_type(4)));
typedef int   v4i __attribute__((ext_vector_type(4)));

typedef __attribute__((address_space(1))) v4i* gv4i_ptr;  // global (prints as __device__)
typedef __attribute__((address_space(3))) v4i* lv4i_ptr;  // LDS    (prints as __shared__)

#define POLY_P 128   // basis size (fixed by reference)
#define BLK    256   // threads per block = 8 wave32
#define CH     32    // k-chunk staged in LDS per round
#define ST     36    // LDS tile row stride in dwords (36*4=144B: 16B aligned, bank-conflict pad)

#define HAS_ASYNC_LDS __has_builtin(__builtin_amdgcn_global_load_async_to_lds_b128)

__global__ __launch_bounds__(BLK)
void OrthogonalPolynomial_47296179863826_kernel(
    const float* __restrict__ xi,
    const float* __restrict__ alpha,
    const float* __restrict__ beta,
    const float* __restrict__ gam,
    float* __restrict__ out,
    int N)
{
    __shared__ __attribute__((aligned(16))) float s_a[POLY_P];
    __shared__ __attribute__((aligned(16))) float s_b[POLY_P];
    __shared__ __attribute__((aligned(16))) float s_g[POLY_P];
    __shared__ __attribute__((aligned(16))) float s_ig[POLY_P];
    __shared__ __attribute__((aligned(16))) float tile[BLK * ST];

    const int tid = threadIdx.x;

#if HAS_ASYNC_LDS
    // gfx1250 async global->LDS: 3 arrays x 128 floats, 32 lanes x b128 each.
    // Bypasses VGPRs entirely; tracked with ASYNCcnt.
    if (tid < (POLY_P / 4)) {
        __builtin_amdgcn_global_load_async_to_lds_b128(
            (gv4i_ptr)(alpha + 4 * tid), (lv4i_ptr)(s_a + 4 * tid), 0, 0);
        __builtin_amdgcn_global_load_async_to_lds_b128(
            (gv4i_ptr)(beta + 4 * tid), (lv4i_ptr)(s_b + 4 * tid), 0, 0);
        __builtin_amdgcn_global_load_async_to_lds_b128(
            (gv4i_ptr)(gam + 4 * tid), (lv4i_ptr)(s_g + 4 * tid), 0, 0);
    }
#if __has_builtin(__builtin_amdgcn_s_wait_asynccnt)
    __builtin_amdgcn_s_wait_asynccnt(0);
#else
    asm volatile("s_wait_asynccnt 0" ::: "memory");
#endif
    __syncthreads();
    // Precompute 1/gamma once per block so the hot loop has no divides.
    if (tid < POLY_P) s_ig[tid] = 1.0f / s_g[tid];
#else
    if (tid < POLY_P) {
        s_a[tid]  = alpha[tid];
        s_b[tid]  = beta[tid];
        s_ig[tid] = 1.0f / gam[tid];
    }
#endif

    const long long blockStart = (long long)blockIdx.x * BLK;
    const long long gid        = blockStart + tid;
    __syncthreads();

    const float x = (gid < (long long)N) ? xi[gid] : 0.0f;

    // phi_0 = 1/g0 ; phi_1 = (x - a0)/g1
    float phi_prev = s_ig[0];
    float phi      = (x - s_a[0]) * s_ig[1];

    const int wave = tid >> 5;         // 8 waves per block (wave32)
    const int lane = tid & 31;
    const int sub  = lane >> 3;        // 4 rows per wave store
    const int ll4  = (lane & 7) << 2;  // 8 lanes x 4 floats = 128B per row chunk

    for (int kc = 0; kc < POLY_P; kc += CH) {
        // ---- compute phase: 32 k-values per thread into padded LDS row ----
        #pragma unroll
        for (int kk = 0; kk < CH; ++kk) {
            const int k = kc + kk;
            float val;
            if (k == 0) {
                val = phi_prev;
            } else if (k == 1) {
                val = phi;
            } else {
                // phi_k = ((x - a[k-1])*phi_{k-1} - b[k-1]*phi_{k-2}) * (1/g[k])
                const float nxt =
                    fmaf(x - s_a[k - 1], phi, -(s_b[k - 1] * phi_prev)) * s_ig[k];
                phi_prev = phi;
                phi      = nxt;
                val      = nxt;
            }
            tile[tid * ST + kk] = val;
        }
        __syncthreads();

        // ---- write phase: each wave drains its own 32 rows ----
        // One wave store = 4 rows x 128B contiguous, cacheline-aligned segments
        // (512B of payload per global_store_b128 issue), nontemporal: the data
        // is streamed once, never re-read.
        #pragma unroll
        for (int r4 = 0; r4 < 8; ++r4) {
            const int row    = (wave << 5) + (r4 << 2) + sub;
            const long long g = blockStart + row;
            if (g < (long long)N) {
                const v4f v = *(const v4f*)&tile[row * ST + ll4];
                __builtin_nontemporal_store(
                    v, (v4f*)(out + (size_t)g * POLY_P + kc + ll4));
            }
        }
        __syncthreads();
    }
}

extern "C" void kernel_launch(void* const* d_in, const int* in_sizes, int n_in,
                              void* d_out, int out_size, void* d_ws, size_t ws_size,
                              hipStream_t stream) {
    const float* xi    = (const float*)d_in[0];
    const float* alpha = (const float*)d_in[1];
    const float* beta  = (const float*)d_in[2];
    const float* gam   = (const float*)d_in[3];
    float* out = (float*)d_out;

    const int N = in_sizes[0];
    const int grid = (N + BLK - 1) / BLK;

    OrthogonalPolynomial_47296179863826_kernel<<<grid, BLK, 0, stream>>>(
        xi, alpha, beta, gam, out, N);
}